// roi_feature_26938034880582
// MI455X (gfx1250) — compile-verified
//
#include <hip/hip_runtime.h>

typedef __attribute__((ext_vector_type(2))) float v2f;
typedef __attribute__((ext_vector_type(8))) float v8f;

#define BIN  16
#define NB   32
#define NC   256
#define NH   64
#define NW   64
#define PAD  66      // padded LDS row stride (floats) to avoid 64-bank conflicts
#define WPB  8       // waves per block (wave32)
#define IMG  (NH*NW)

// out[b,c] = Wy[b] (16x64) * F[b,c] (64x64) * Wx[b]^T (64x16), per tensor.
__global__ __launch_bounds__(256) void roi_bilinear_wmma_kernel(
    const float* __restrict__ feat0,   // original_features (32,256,64,64)
    const float* __restrict__ lab,     // (32,1,5)
    const float* __restrict__ feat1,   // attack_features  (32,256,64,64)
    float* __restrict__ out)           // (2,32,256,16,16) flattened
{
  __shared__ float sWy[BIN * PAD];
  __shared__ float sWx[BIN * PAD];
  __shared__ float sS[WPB][BIN * PAD];   // per-wave intermediate S = Wy*F (16x64)

  const int b    = blockIdx.y;
  const int cgrp = blockIdx.x;
  const int tid  = threadIdx.x;
  const int lane = tid & 31;
  const int wave = tid >> 5;

  // ---- build sparse interpolation matrices Wy (rows j) / Wx (rows k) ----
  for (int i = tid; i < BIN * PAD; i += 256) { sWy[i] = 0.0f; sWx[i] = 0.0f; }
  __syncthreads();

  const float* labb = lab + b * 5;
  if (tid < BIN) {
    // y-coordinate: reference grid[...,1] = mesh_x -> lab[4], lab[2]
    int j = tid;
    float ynorm = 2.0f * (float)(j - BIN / 2) * labb[4] / (float)BIN + (labb[2] * 2.0f - 1.0f);
    float yc  = ((ynorm + 1.0f) * (float)NH - 1.0f) * 0.5f;
    float y0f = floorf(yc);
    int   y0  = (int)y0f;
    float w1  = yc - y0f, w0 = 1.0f - w1;
    if (y0 >= 0     && y0 < NH)     sWy[j * PAD + y0]     = w0;
    if (y0 + 1 >= 0 && y0 + 1 < NH) sWy[j * PAD + y0 + 1] = w1;
  } else if (tid < 2 * BIN) {
    // x-coordinate: reference grid[...,0] = mesh_y -> lab[3], lab[1]
    int k = tid - BIN;
    float xnorm = 2.0f * (float)(k - BIN / 2) * labb[3] / (float)BIN + (labb[1] * 2.0f - 1.0f);
    float xc  = ((xnorm + 1.0f) * (float)NW - 1.0f) * 0.5f;
    float x0f = floorf(xc);
    int   x0  = (int)x0f;
    float w1  = xc - x0f, w0 = 1.0f - w1;
    if (x0 >= 0     && x0 < NW)     sWx[k * PAD + x0]     = w0;
    if (x0 + 1 >= 0 && x0 + 1 < NW) sWx[k * PAD + x0 + 1] = w1;
  }
  __syncthreads();

  const int c     = cgrp * WPB + wave;     // one channel per wave
  const int row16 = lane & 15;             // M/N index within fragment
  const int koff  = (lane >> 4) << 1;      // lanes 16-31 hold K+2,K+3
  float* Sbuf = sS[wave];

  const float* feats[2] = { feat0, feat1 };
  #pragma unroll
  for (int t = 0; t < 2; ++t) {
    const float* F = feats[t] + (size_t)(b * NC + c) * IMG;

    // ---- Stage 1: S (16x64) = Wy (16x64) x F (64x64), K = y ----
    for (int n = 0; n < 4; ++n) {          // 16-wide N tiles over x
      v8f acc = {};
      #pragma unroll
      for (int k = 0; k < 16; ++k) {       // K steps of 4
        // A fragment: Wy[row16][4k+koff .. +1]  (LDS, bank-conflict-free)
        const float* ap = sWy + row16 * PAD + 4 * k + koff;
        v2f a; a.x = ap[0]; a.y = ap[1];
        // B fragment: F rows (4k+koff, 4k+koff+1), cols 16n..16n+15 (coalesced global)
        const float* bp = F + (4 * k + koff) * NW + n * 16 + row16;
        v2f bm; bm.x = bp[0]; bm.y = bp[NW];
        acc = __builtin_amdgcn_wmma_f32_16x16x4_f32(false, a, false, bm,
                                                    (short)0, acc, false, false);
      }
      // spill S tile (D layout) -> row-major padded LDS (same-wave, in-order DS)
      #pragma unroll
      for (int j2 = 0; j2 < 8; ++j2) {
        int r = (lane < 16) ? j2 : (j2 + 8);
        Sbuf[r * PAD + n * 16 + row16] = acc[j2];
      }
    }

    // ---- Stage 2: D (16x16) = S (16x64) x Wx^T (64x16), K = x ----
    v8f d = {};
    #pragma unroll
    for (int k = 0; k < 16; ++k) {
      const float* ap = Sbuf + row16 * PAD + 4 * k + koff;   // A from S
      v2f a; a.x = ap[0]; a.y = ap[1];
      const float* bp = sWx + row16 * PAD + 4 * k + koff;    // B[kx][gx] = Wx[gx][kx]
      v2f bm; bm.x = bp[0]; bm.y = bp[1];
      d = __builtin_amdgcn_wmma_f32_16x16x4_f32(false, a, false, bm,
                                                (short)0, d, false, false);
    }

    // ---- store D tile: out[t][b][c][gy][gx], coalesced 64B per half-wave ----
    size_t base = (size_t)t * (size_t)(NB * NC * BIN * BIN)
                + (size_t)(b * NC + c) * (BIN * BIN);
    #pragma unroll
    for (int j2 = 0; j2 < 8; ++j2) {
      int r = (lane < 16) ? j2 : (j2 + 8);
      out[base + r * BIN + row16] = d[j2];
    }
  }
}

extern "C" void kernel_launch(void* const* d_in, const int* in_sizes, int n_in,
                              void* d_out, int out_size, void* d_ws, size_t ws_size,
                              hipStream_t stream) {
  const float* orig = (const float*)d_in[0];   // original_features
  const float* lab  = (const float*)d_in[1];   // lab
  const float* atk  = (const float*)d_in[2];   // attack_features
  float* out = (float*)d_out;

  dim3 grid(NC / WPB, NB);   // (32 channel-groups, 32 batches)
  roi_bilinear_wmma_kernel<<<grid, 256, 0, stream>>>(orig, lab, atk, out);
}